// GaussianVectorQuantizer_1279900254455
// MI455X (gfx1250) — compile-verified
//
#include <hip/hip_runtime.h>
#include <hip/hip_bf16.h>

// Gaussian Vector Quantizer forward for MI455X (gfx1250, wave32, WMMA).
// D = A*B+C via V_WMMA_F32_16X16X4_F32 for both GEMMs (f32 fidelity).
// Each wave computes two 16x16 output tiles per K sweep, sharing one A
// fragment -> 2 WMMA per ds_load_2addr_b64, global B via b64 loads only.

typedef __attribute__((ext_vector_type(2))) float v2f;
typedef __attribute__((ext_vector_type(8))) float v8f;

#define EPSQ 1e-10f

__device__ __forceinline__ v8f wmma_f32(v2f a, v2f b, v8f c) {
  // 8 args: (neg_a, A, neg_b, B, c_mod, C, reuse_a, reuse_b)
  return __builtin_amdgcn_wmma_f32_16x16x4_f32(false, a, false, b, (short)0, c,
                                               false, false);
}

__device__ __forceinline__ unsigned rotl32(unsigned x, unsigned d) {
  return (x << d) | (x >> (32u - d));
}

// Exact JAX threefry2x32, key = jax.random.key(42) -> (ks0, ks1) = (0, 42).
// JAX generates bits over iota(total) split in halves: out[i<half] = x0(i, i+half),
// out[i>=half] = x1(i-half, i). uniform = bitcast((bits>>9)|0x3f800000) - 1.
__device__ __forceinline__ float jax_uniform(unsigned idx, unsigned halfT) {
  unsigned c0, c1;
  bool hi;
  if (idx < halfT) { c0 = idx;         c1 = idx + halfT; hi = false; }
  else             { c0 = idx - halfT; c1 = idx;         hi = true;  }
  const unsigned ks0 = 0u, ks1 = 42u, ks2 = 0x1BD11BDAu ^ 0u ^ 42u;
  unsigned x0 = c0 + ks0, x1 = c1 + ks1;
#define TF_R(r) { x0 += x1; x1 = rotl32(x1, r); x1 ^= x0; }
  TF_R(13) TF_R(15) TF_R(26) TF_R(6)   x0 += ks1; x1 += ks2 + 1u;
  TF_R(17) TF_R(29) TF_R(16) TF_R(24)  x0 += ks2; x1 += ks0 + 2u;
  TF_R(13) TF_R(15) TF_R(26) TF_R(6)   x0 += ks0; x1 += ks1 + 3u;
  TF_R(17) TF_R(29) TF_R(16) TF_R(24)  x0 += ks1; x1 += ks2 + 4u;
  TF_R(13) TF_R(15) TF_R(26) TF_R(6)   x0 += ks2; x1 += ks0 + 5u;
#undef TF_R
  unsigned bits = hi ? x1 : x0;
  return __uint_as_float((bits >> 9) | 0x3F800000u) - 1.0f;
}

// ---- prep: book row norms (K rows of 256) + precision_q scalar ----
__global__ __launch_bounds__(256) void gvq_prep_kernel(
    const float* __restrict__ book, const float* __restrict__ lpq,
    float* __restrict__ bn, float* __restrict__ out_prec) {
  __shared__ float red[256];
  const int row = blockIdx.x, tid = threadIdx.x;
  float v = book[(size_t)row * 256 + tid];
  red[tid] = v * v;
  __syncthreads();
  for (int s = 128; s > 0; s >>= 1) {
    if (tid < s) red[tid] += red[tid + s];
    __syncthreads();
  }
  if (tid == 0) {
    bn[row] = red[0];
    if (row == 0) *out_prec = 0.5f / fmaxf(__expf(lpq[0]), EPSQ);
  }
}

// ---- prep: bookT[256][1024] = transpose(book[1024][256]) (tiled via LDS) ----
__global__ void gvq_transpose_kernel(const float* __restrict__ in,   // [1024][256]
                                     float* __restrict__ out) {      // [256][1024]
  __shared__ float t[32][33];
  const int bx = blockIdx.x * 32;  // column (dim) base
  const int by = blockIdx.y * 32;  // row (code) base
  const int x = threadIdx.x, y = threadIdx.y;  // blockDim = (32, 8)
#pragma unroll
  for (int i = 0; i < 32; i += 8) t[y + i][x] = in[(size_t)(by + y + i) * 256 + bx + x];
  __syncthreads();
#pragma unroll
  for (int i = 0; i < 32; i += 8) out[(size_t)(bx + y + i) * 1024 + by + x] = t[x][y + i];
}

// ---- fused main kernel: 16 rows per block, 256 threads (8 wave32) ----
// BT = true: GEMM2 reads contiguous bookT rows (b64 loads); else strided book.
template <bool BT>
__global__ __launch_bounds__(256) void gvq_kernel(
    const float* __restrict__ z, const float* __restrict__ book,
    const float* __restrict__ bookT, const float* __restrict__ lpq,
    const float* __restrict__ temp, const int* __restrict__ is_train,
    const float* __restrict__ bn, float* __restrict__ out_zq,
    float* __restrict__ out_logits) {
  constexpr int DIM = 256, KK = 1024, BR = 16;
  constexpr int LZP = DIM + 4;  // +4 pad: row stride % 64 banks == 4 -> conflict-free
  constexpr int LPP = KK + 4;
  __shared__ float lz[BR][LZP];   // ~16.6 KB
  __shared__ float lzn[BR];
  __shared__ float lP[BR][LPP];   // ~65.8 KB (logits, then probabilities)
  __shared__ int iw[BR];

  const int tid = threadIdx.x;
  const int wave = tid >> 5, lane = tid & 31;
  const int lm = lane & 15;                // M (or N) index within a 16-wide tile
  const int koff = (lane & 16) ? 2 : 0;    // A/B: upper lane half holds K+2, K+3
  const int mhi = (lane & 16) ? 8 : 0;     // C/D: upper lane half holds M+8
  const int r0 = blockIdx.x * BR;

  const float prec = 0.5f / fmaxf(__expf(lpq[0]), EPSQ);
  const float invT = 1.0f / temp[0];
  const bool train = (is_train[0] != 0);

  // Stage z rows into LDS (coalesced float4 loads).
  {
    const float4* zsrc = (const float4*)(z + (size_t)r0 * DIM);
    for (int i = tid; i < BR * DIM / 4; i += 256) {
      int row = i / (DIM / 4), c4 = i % (DIM / 4);
      *(float4*)&lz[row][c4 * 4] = zsrc[i];
    }
  }
  __syncthreads();
  if (tid < BR) {  // deterministic per-row ||z||^2
    float s = 0.f;
    for (int k = 0; k < DIM; k++) { float v = lz[tid][k]; s += v * v; }
    lzn[tid] = s;
  }
  __syncthreads();

  // ---- GEMM1: logits[16,1024] = -(||z||^2 + ||c||^2 - 2 z.book^T) * prec ----
  // 64 column tiles as 32 pairs; each wave owns 4 pairs. One shared A fragment
  // feeds two V_WMMA_F32_16X16X4_F32 per k-step.
  for (int tp = wave; tp < KK / 32; tp += 8) {
    const int n0a = tp * 16;
    const int n0b = (tp + KK / 32) * 16;
    const float* browa = book + (size_t)(n0a + lm) * DIM + koff;  // B[k][n]=book[n][k]
    const float* browb = book + (size_t)(n0b + lm) * DIM + koff;
    v8f acca = {0.f, 0.f, 0.f, 0.f, 0.f, 0.f, 0.f, 0.f};
    v8f accb = {0.f, 0.f, 0.f, 0.f, 0.f, 0.f, 0.f, 0.f};
#pragma unroll 4
    for (int k = 0; k < DIM; k += 4) {
      v2f a = *(const v2f*)&lz[lm][k + koff];  // A 16x4: lanes 0-15 K,K+1; 16-31 K+2,K+3
      acca = wmma_f32(a, *(const v2f*)&browa[k], acca);
      accb = wmma_f32(a, *(const v2f*)&browb[k], accb);
    }
    const float bnva = bn[n0a + lm];
    const float bnvb = bn[n0b + lm];
#pragma unroll
    for (int i = 0; i < 8; i++) {
      const int m = i + mhi;  // C/D: VGPR i -> M=i (lanes 0-15) / M=i+8 (lanes 16-31)
      float lga = -(lzn[m] + bnva - 2.0f * acca[i]) * prec;
      float lgb = -(lzn[m] + bnvb - 2.0f * accb[i]) * prec;
      lP[m][n0a + lm] = lga;
      lP[m][n0b + lm] = lgb;
      out_logits[(size_t)(r0 + m) * KK + n0a + lm] = lga;
      out_logits[(size_t)(r0 + m) * KK + n0b + lm] = lgb;
    }
  }
  __syncthreads();

  if (train) {
    // ---- Gumbel-softmax over K=1024, 16 lanes per row, in-place in LDS ----
    const int row = tid >> 4, j = tid & 15;
    const unsigned grow = (unsigned)(r0 + row);
    const unsigned halfT = (unsigned)gridDim.x * (unsigned)BR * (unsigned)(KK / 2);
    float mx = -3.4e38f;
    for (int c = j; c < KK; c += 16) {
      float u = jax_uniform(grow * (unsigned)KK + (unsigned)c, halfT);
      float g = -__logf(-__logf(u + EPSQ) + EPSQ);
      float s = (lP[row][c] + g) * invT;
      lP[row][c] = s;
      mx = fmaxf(mx, s);
    }
    for (int m2 = 1; m2 < 16; m2 <<= 1) mx = fmaxf(mx, __shfl_xor(mx, m2, 16));
    float sum = 0.f;
    for (int c = j; c < KK; c += 16) {
      float e = __expf(lP[row][c] - mx);
      lP[row][c] = e;
      sum += e;
    }
    for (int m2 = 1; m2 < 16; m2 <<= 1) sum += __shfl_xor(sum, m2, 16);
    const float inv = 1.0f / sum;
    for (int c = j; c < KK; c += 16) lP[row][c] *= inv;
    __syncthreads();

    // ---- GEMM2: zq[16,256] = P[16,1024] @ book[1024,256] ----
    // 16 column tiles = exactly 2 per wave; shared A fragment, 2 WMMA/k-step.
    {
      const int d0a = wave * 16;
      const int d0b = (wave + 8) * 16;
      v8f acca = {0.f, 0.f, 0.f, 0.f, 0.f, 0.f, 0.f, 0.f};
      v8f accb = {0.f, 0.f, 0.f, 0.f, 0.f, 0.f, 0.f, 0.f};
      if (BT) {
        // bookT[d][k]: contiguous along k -> b64 loads like GEMM1.
        const float* bra = bookT + (size_t)(d0a + lm) * KK + koff;
        const float* brb = bookT + (size_t)(d0b + lm) * KK + koff;
#pragma unroll 4
        for (int k = 0; k < KK; k += 4) {
          v2f a = *(const v2f*)&lP[lm][k + koff];
          acca = wmma_f32(a, *(const v2f*)&bra[k], acca);
          accb = wmma_f32(a, *(const v2f*)&brb[k], accb);
        }
      } else {
        const float* bca = book + (size_t)koff * DIM + d0a + lm;
        const float* bcb = book + (size_t)koff * DIM + d0b + lm;
#pragma unroll 4
        for (int k = 0; k < KK; k += 4) {
          v2f a = *(const v2f*)&lP[lm][k + koff];
          v2f ba, bb;
          ba.x = bca[(size_t)k * DIM];
          ba.y = bca[(size_t)k * DIM + DIM];
          bb.x = bcb[(size_t)k * DIM];
          bb.y = bcb[(size_t)k * DIM + DIM];
          acca = wmma_f32(a, ba, acca);
          accb = wmma_f32(a, bb, accb);
        }
      }
#pragma unroll
      for (int i = 0; i < 8; i++) {
        const int m = i + mhi;
        out_zq[(size_t)(r0 + m) * DIM + d0a + lm] = acca[i];
        out_zq[(size_t)(r0 + m) * DIM + d0b + lm] = accb[i];
      }
    }
  } else {
    // ---- eval: row argmax (first max on ties, like jnp.argmax) + gather ----
    const int row = tid >> 4, j = tid & 15;
    float bv = -3.4e38f;
    int bi = 0x7fffffff;
    for (int c = j; c < KK; c += 16) {
      float v = lP[row][c];
      if (v > bv || (v == bv && c < bi)) { bv = v; bi = c; }
    }
    for (int m2 = 1; m2 < 16; m2 <<= 1) {
      float ov = __shfl_xor(bv, m2, 16);
      int oi = __shfl_xor(bi, m2, 16);
      if (ov > bv || (ov == bv && oi < bi)) { bv = ov; bi = oi; }
    }
    if (j == 0) iw[row] = bi;
    __syncthreads();
    for (int i = tid; i < BR * DIM; i += 256) {
      int row2 = i / DIM, d = i % DIM;
      out_zq[(size_t)(r0 + row2) * DIM + d] = book[(size_t)iw[row2] * DIM + d];
    }
  }
}

extern "C" void kernel_launch(void* const* d_in, const int* in_sizes, int n_in,
                              void* d_out, int out_size, void* d_ws, size_t ws_size,
                              hipStream_t stream) {
  const float* z    = (const float*)d_in[0];
  const float* lpq  = (const float*)d_in[1];
  const float* temp = (const float*)d_in[2];
  const int* istr   = (const int*)d_in[3];
  const float* book = (const float*)d_in[4];
  float* out = (float*)d_out;
  float* ws = (float*)d_ws;

  const int DIM = 256;
  const int N = in_sizes[0] / DIM;   // 65536 flattened z rows
  const int K = in_sizes[4] / DIM;   // 1024 codebook entries
  const size_t zq_elems = (size_t)N * DIM;

  float* bn = ws;                    // K floats
  float* bookT = ws + K;             // DIM*K floats (if workspace allows)
  const bool has_bt = ws_size >= (size_t)(K + DIM * K) * sizeof(float);

  float* out_zq = out;
  float* out_prec = out + zq_elems;
  float* out_logits = out + zq_elems + 1;

  gvq_prep_kernel<<<dim3(K), dim3(256), 0, stream>>>(book, lpq, bn, out_prec);
  if (has_bt) {
    gvq_transpose_kernel<<<dim3(DIM / 32, K / 32), dim3(32, 8), 0, stream>>>(book, bookT);
    gvq_kernel<true><<<dim3(N / 16), dim3(256), 0, stream>>>(
        z, book, bookT, lpq, temp, istr, bn, out_zq, out_logits);
  } else {
    gvq_kernel<false><<<dim3(N / 16), dim3(256), 0, stream>>>(
        z, book, bookT, lpq, temp, istr, bn, out_zq, out_logits);
  }
}